// FieldAwareFactorizationMachine_41609643164216
// MI455X (gfx1250) — compile-verified
//
#include <hip/hip_runtime.h>
#include <math.h>

typedef __attribute__((ext_vector_type(2))) float v2f;
typedef __attribute__((ext_vector_type(8))) float v8f;

#define NFIELD 20
#define FDIM   50000
#define VTOT   1000000        // 20 * 50000
#define EMB    16
#define NPAIR  190            // 20*19/2
#define NTILE  12             // ceil(190/16)
#define WAVES_PER_BLOCK 4

// Packed (i | j<<8) upper-triangle pair table, padded to 192 entries.
struct PairTab { short v[NTILE * 16]; };
static constexpr PairTab make_pairs() {
  PairTab t{};
  int p = 0;
  for (int i = 0; i < NFIELD; ++i)
    for (int j = i + 1; j < NFIELD; ++j)
      t.v[p++] = (short)(i | (j << 8));
  while (p < NTILE * 16) t.v[p++] = 0;
  return t;
}
__constant__ PairTab g_pairs = make_pairs();

__global__ __launch_bounds__(WAVES_PER_BLOCK * 32)
void ffm_wmma_kernel(const int* __restrict__ x,
                     const float* __restrict__ Wl,
                     const float* __restrict__ Wc,
                     const float* __restrict__ bias,
                     float* __restrict__ out,
                     int nbatch)
{
  __shared__ int xo_s[WAVES_PER_BLOCK][NFIELD];

  const int tid    = threadIdx.x;
  const int w      = tid >> 5;
  const int lane   = tid & 31;
  const int lo     = lane & 15;   // pair slot within tile / N column
  const int hi     = lane >> 4;   // K half selector
  const int sample = blockIdx.x * WAVES_PER_BLOCK + w;

  if (sample < nbatch && lane < NFIELD) {
    // xo[b,f] = x[b,f] + f * 50000
    xo_s[w][lane] = x[sample * NFIELD + lane] + lane * FDIM;
  }
  __syncthreads();
  if (sample >= nbatch) return;   // whole-wave uniform exit

  v8f acc0 = {};
  v8f acc1 = {};
  const int eb = hi * 2;          // element offset base within each K=4 chunk

  for (int t = 0; t < NTILE; ++t) {
    const int p = t * 16 + lo;
    v2f a0 = {}, a1 = {}, a2 = {}, a3 = {};
    v2f b0 = {}, b1 = {}, b2 = {}, b3 = {};
    if (p < NPAIR) {
      const int pr = (int)g_pairs.v[p];
      const int fi = pr & 0xff;
      const int fj = pr >> 8;
      const unsigned xi = (unsigned)xo_s[w][fi];   // xo[b, i]
      const unsigned xj = (unsigned)xo_s[w][fj];   // xo[b, j]
      // a_vec = W_cross[j, xo_i, :] ; b_vec = W_cross[i, xo_j, :]
      const float* ap = Wc + ((unsigned)fj * (unsigned)VTOT + xi) * EMB + eb;
      const float* bp = Wc + ((unsigned)fi * (unsigned)VTOT + xj) * EMB + eb;
      a0 = *(const v2f*)(ap + 0);
      a1 = *(const v2f*)(ap + 4);
      a2 = *(const v2f*)(ap + 8);
      a3 = *(const v2f*)(ap + 12);
      b0 = *(const v2f*)(bp + 0);
      b1 = *(const v2f*)(bp + 4);
      b2 = *(const v2f*)(bp + 8);
      b3 = *(const v2f*)(bp + 12);
    }
    // Four chained K=4 f32 WMMAs: acc += A_chunk x B_chunk (exact f32 FMA)
    acc0 = __builtin_amdgcn_wmma_f32_16x16x4_f32(false, a0, false, b0,
                                                 (short)0, acc0, false, false);
    acc1 = __builtin_amdgcn_wmma_f32_16x16x4_f32(false, a1, false, b1,
                                                 (short)0, acc1, false, false);
    acc0 = __builtin_amdgcn_wmma_f32_16x16x4_f32(false, a2, false, b2,
                                                 (short)0, acc0, false, false);
    acc1 = __builtin_amdgcn_wmma_f32_16x16x4_f32(false, a3, false, b3,
                                                 (short)0, acc1, false, false);
  }

  // Extract diagonal of the 16x16 D tile:
  //   D[m][m] for m=0..7  lives in C[k] on lanes 0..7   (lane == k)
  //   D[m][m] for m=8..15 lives in C[k] on lanes 24..31 (lane == k+24)
  float diag = 0.f;
#pragma unroll
  for (int k = 0; k < 8; ++k) {
    const float v = acc0[k] + acc1[k];
    if (lane == k || lane == k + 24) diag = v;
  }

  // Linear term: lanes 0..19 each fetch one W_linear entry.
  float lin = 0.f;
  if (lane < NFIELD) lin = Wl[(unsigned)xo_s[w][lane]];

  // Wave32 tree reduction of (diagonal dots + linear term).
  float tot = diag + lin;
#pragma unroll
  for (int off = 16; off > 0; off >>= 1)
    tot += __shfl_xor(tot, off, 32);

  if (lane == 0) {
    const float z = tot + bias[0];
    out[sample] = 1.0f / (1.0f + expf(-z));
  }
}

extern "C" void kernel_launch(void* const* d_in, const int* in_sizes, int n_in,
                              void* d_out, int out_size, void* d_ws, size_t ws_size,
                              hipStream_t stream) {
  const int*   x   = (const int*)d_in[0];     // (4096, 20) int32
  const float* Wl  = (const float*)d_in[1];   // (1e6, 1) f32
  const float* Wc  = (const float*)d_in[2];   // (20, 1e6, 16) f32
  const float* bs  = (const float*)d_in[3];   // (1,) f32
  float*       out = (float*)d_out;           // (4096,) f32

  const int nbatch = out_size;
  const int blocks = (nbatch + WAVES_PER_BLOCK - 1) / WAVES_PER_BLOCK;
  hipLaunchKernelGGL(ffm_wmma_kernel, dim3(blocks), dim3(WAVES_PER_BLOCK * 32),
                     0, stream, x, Wl, Wc, bs, out, nbatch);
}